// TemporalGCN_85727547228377
// MI455X (gfx1250) — compile-verified
//
#include <hip/hip_runtime.h>
#include <hip/hip_bf16.h>

typedef __attribute__((ext_vector_type(2))) float v2f;
typedef __attribute__((ext_vector_type(8))) float v8f;

#define NN 20000
#define TT 12
#define FF 64
#define HH 64
#define LL 64
#define EE 640000
#define MTILES 1250  // NN / 16

__device__ __forceinline__ float sigf(float x) { return 1.0f / (1.0f + expf(-x)); }

// ---- init: h = c = 0, deg = 0 ----------------------------------------------
__global__ void k_init(float* __restrict__ h, float* __restrict__ c, float* __restrict__ deg) {
  int i = blockIdx.x * blockDim.x + threadIdx.x;
  if (i < NN * LL) { h[i] = 0.0f; c[i] = 0.0f; }
  if (i < NN) deg[i] = 0.0f;
}

// ---- in-degree over dst (self-loop added later as +1) ----------------------
__global__ void k_deg(const int* __restrict__ ei, float* __restrict__ deg) {
  int e = blockIdx.x * blockDim.x + threadIdx.x;
  if (e < EE) atomicAdd(&deg[ei[EE + e]], 1.0f);
}

__global__ void k_dinv(const float* __restrict__ deg, float* __restrict__ dinv) {
  int n = blockIdx.x * blockDim.x + threadIdx.x;
  if (n < NN) dinv[n] = rsqrtf(deg[n] + 1.0f);  // self-loop guarantees deg>0
}

// ---- combined gate bias: b_ih + b_hh + gcn_bias @ w_ih^T -------------------
__global__ void k_bias(const float* __restrict__ b_ih, const float* __restrict__ b_hh,
                       const float* __restrict__ gb, const float* __restrict__ w_ih,
                       float* __restrict__ bcomb) {
  int g = threadIdx.x;  // 256 threads
  float s = b_ih[g] + b_hh[g];
  for (int k = 0; k < HH; ++k) s += gb[k] * w_ih[g * HH + k];
  bcomb[g] = s;
}

// ---- GCN GEMM: xw = x_t @ gcn_weight^T  (20000x64 @ 64x64) -----------------
// One wave per 16-row tile; gcn_weight staged once per block into LDS (16 KB),
// B fragments fed from ds_load. WMMA f32 16x16x4, K-loop of 16.
__global__ void k_gcn_mm(const float* __restrict__ xt, const float* __restrict__ gw,
                         float* __restrict__ xw) {
  __shared__ float sgw[FF * HH];  // 4096 floats = 16 KB
  int tid = threadIdx.x;          // 128 threads
#pragma unroll
  for (int i = 0; i < 8; ++i) {
    int off = (i * 128 + tid) * 4;
    *(float4*)(sgw + off) = *(const float4*)(gw + off);
  }
  __syncthreads();

  int tile = blockIdx.x * 4 + (tid >> 5);
  if (tile >= MTILES) return;  // wave-uniform; EXEC stays all-ones for WMMA
  int lane = tid & 31;
  int mL = lane & 15, hi = lane >> 4;
  int rowbase = tile * 16;

  v2f a[16];
#pragma unroll
  for (int s = 0; s < 16; ++s)
    a[s] = *(const v2f*)(xt + (size_t)(rowbase + mL) * FF + 4 * s + 2 * hi);

#pragma unroll
  for (int jt = 0; jt < 4; ++jt) {
    v8f acc = {};
#pragma unroll
    for (int s = 0; s < 16; ++s) {
      // B[k][j] = W[j][k]  (C = X @ W^T), W tile from LDS
      v2f b = *(const v2f*)(sgw + (jt * 16 + mL) * FF + 4 * s + 2 * hi);
      acc = __builtin_amdgcn_wmma_f32_16x16x4_f32(false, a[s], false, b, (short)0, acc,
                                                  false, false);
    }
#pragma unroll
    for (int v = 0; v < 8; ++v)
      xw[(size_t)(rowbase + v + 8 * hi) * HH + jt * 16 + mL] = acc[v];
  }
}

// ---- self-loop term initializes agg: agg[n] = dinv[n]^2 * xw[n] ------------
__global__ void k_selfloop(const float* __restrict__ xw, const float* __restrict__ dinv,
                           float* __restrict__ agg) {
  int i = blockIdx.x * blockDim.x + threadIdx.x;
  if (i < NN * HH) {
    int n = i >> 6;
    float d = dinv[n];
    agg[i] = d * d * xw[i];
  }
}

// ---- edge scatter: agg[dst] += dinv[src]*dinv[dst] * xw[src] ---------------
// One lane per (edge, feature-pair); L2-resident f32 atomics.
__global__ void k_edge(const int* __restrict__ ei, const float* __restrict__ dinv,
                       const float* __restrict__ xw, float* __restrict__ agg) {
  int gid = blockIdx.x * blockDim.x + threadIdx.x;
  int e = gid >> 5;
  if (e >= EE) return;
  int j2 = (gid & 31) * 2;
  int s = ei[e], d = ei[EE + e];
  float nrm = dinv[s] * dinv[d];
  v2f v = *(const v2f*)(xw + (size_t)s * HH + j2);
  atomicAdd(&agg[(size_t)d * HH + j2], nrm * v.x);
  atomicAdd(&agg[(size_t)d * HH + j2 + 1], nrm * v.y);
}

// ---- fused LSTM step: gates = agg@w_ih^T + h@w_hh^T + bcomb; update h,c ----
// Both weight matrices (128 KB) staged once per block into dynamic LDS; one
// wave per 16-node tile. A fragments (agg + h rows) cached in registers and
// reused across all 16 gate-column tiles. i/f/g/o for a given (node,l) land in
// matching lane/element positions across the 4 chunk accumulators, so the
// nonlinear update is pure-register (no cross-wave traffic). h/c row-local.
__global__ void k_lstm(const float* __restrict__ agg, float* __restrict__ h,
                       float* __restrict__ c, const float* __restrict__ w_ih,
                       const float* __restrict__ w_hh, const float* __restrict__ bcomb) {
  extern __shared__ float smw[];  // [0,16384): w_ih  [16384,32768): w_hh
  int tid = threadIdx.x;          // 128 threads
#pragma unroll 4
  for (int i = 0; i < 32; ++i) {
    int off = (i * 128 + tid) * 4;
    *(float4*)(smw + off) = *(const float4*)(w_ih + off);
  }
#pragma unroll 4
  for (int i = 0; i < 32; ++i) {
    int off = (i * 128 + tid) * 4;
    *(float4*)(smw + 16384 + off) = *(const float4*)(w_hh + off);
  }
  __syncthreads();

  int tile = blockIdx.x * 4 + (tid >> 5);
  if (tile >= MTILES) return;  // wave-uniform; EXEC all-ones for WMMA
  int lane = tid & 31;
  int mL = lane & 15, hi = lane >> 4;
  int rowbase = tile * 16;

  v2f aA[16], aH[16];
#pragma unroll
  for (int s = 0; s < 16; ++s) {
    aA[s] = *(const v2f*)(agg + (size_t)(rowbase + mL) * HH + 4 * s + 2 * hi);
    aH[s] = *(const v2f*)(h + (size_t)(rowbase + mL) * LL + 4 * s + 2 * hi);
  }

#pragma unroll
  for (int nt = 0; nt < 4; ++nt) {
    v8f acc[4];
#pragma unroll
    for (int ch = 0; ch < 4; ++ch) {  // gate blocks: i, f, g, o
      int jcol = ch * 64 + nt * 16 + mL;
      v8f t = {};
#pragma unroll
      for (int s = 0; s < 16; ++s) {
        v2f b = *(const v2f*)(smw + jcol * HH + 4 * s + 2 * hi);
        t = __builtin_amdgcn_wmma_f32_16x16x4_f32(false, aA[s], false, b, (short)0, t,
                                                  false, false);
      }
#pragma unroll
      for (int s = 0; s < 16; ++s) {
        v2f b = *(const v2f*)(smw + 16384 + jcol * LL + 4 * s + 2 * hi);
        t = __builtin_amdgcn_wmma_f32_16x16x4_f32(false, aH[s], false, b, (short)0, t,
                                                  false, false);
      }
      acc[ch] = t;
    }
    float b0 = bcomb[0 * 64 + nt * 16 + mL];
    float b1 = bcomb[1 * 64 + nt * 16 + mL];
    float b2 = bcomb[2 * 64 + nt * 16 + mL];
    float b3 = bcomb[3 * 64 + nt * 16 + mL];
#pragma unroll
    for (int v = 0; v < 8; ++v) {
      size_t idx = (size_t)(rowbase + v + 8 * hi) * LL + nt * 16 + mL;
      float gi = acc[0][v] + b0;
      float gf = acc[1][v] + b1;
      float gg = acc[2][v] + b2;
      float go = acc[3][v] + b3;
      float cv = sigf(gf) * c[idx] + sigf(gi) * tanhf(gg);
      c[idx] = cv;
      h[idx] = sigf(go) * tanhf(cv);
    }
  }
}

// ---- output: out[n] = h[n] . lin_w + lin_b ---------------------------------
__global__ void k_out(const float* __restrict__ h, const float* __restrict__ lw,
                      const float* __restrict__ lb, float* __restrict__ out) {
  int n = blockIdx.x * blockDim.x + threadIdx.x;
  if (n >= NN) return;
  float s = lb[0];
#pragma unroll
  for (int l = 0; l < LL; ++l) s += h[(size_t)n * LL + l] * lw[l];
  out[n] = s;
}

extern "C" void kernel_launch(void* const* d_in, const int* in_sizes, int n_in,
                              void* d_out, int out_size, void* d_ws, size_t ws_size,
                              hipStream_t stream) {
  const float* x_seq = (const float*)d_in[0];
  const int* ei      = (const int*)d_in[1];
  const float* gw    = (const float*)d_in[2];
  const float* gb    = (const float*)d_in[3];
  const float* w_ih  = (const float*)d_in[4];
  const float* w_hh  = (const float*)d_in[5];
  const float* b_ih  = (const float*)d_in[6];
  const float* b_hh  = (const float*)d_in[7];
  const float* lw    = (const float*)d_in[8];
  const float* lb    = (const float*)d_in[9];
  float* out = (float*)d_out;

  // workspace layout (~21 MB)
  float* ws    = (float*)d_ws;
  float* deg   = ws;                       // NN
  float* dinv  = ws + NN;                  // NN
  float* bcomb = ws + 2 * NN;              // 256
  float* xw    = bcomb + 256;              // NN*HH
  float* agg   = xw + (size_t)NN * HH;     // NN*HH
  float* h     = agg + (size_t)NN * HH;    // NN*LL
  float* c     = h + (size_t)NN * LL;      // NN*LL

  k_init<<<(NN * LL + 255) / 256, 256, 0, stream>>>(h, c, deg);
  k_deg<<<(EE + 255) / 256, 256, 0, stream>>>(ei, deg);
  k_dinv<<<(NN + 255) / 256, 256, 0, stream>>>(deg, dinv);
  k_bias<<<1, 256, 0, stream>>>(b_ih, b_hh, gb, w_ih, bcomb);

  const size_t lstm_lds = 32768 * sizeof(float);  // 128 KB (WGP has 320 KB)
  for (int t = 0; t < TT; ++t) {
    const float* xt = x_seq + (size_t)t * NN * FF;
    k_gcn_mm<<<(MTILES + 3) / 4, 128, 0, stream>>>(xt, gw, xw);
    k_selfloop<<<(NN * HH + 255) / 256, 256, 0, stream>>>(xw, dinv, agg);
    k_edge<<<(EE * 32 + 255) / 256, 256, 0, stream>>>(ei, dinv, xw, agg);
    k_lstm<<<(MTILES + 3) / 4, 128, lstm_lds, stream>>>(agg, h, c, w_ih, w_hh, bcomb);
  }
  k_out<<<(NN + 255) / 256, 256, 0, stream>>>(h, lw, lb, out);
}